// MultimodalRetrieval_74448963109447
// MI455X (gfx1250) — compile-verified
//
#include <hip/hip_runtime.h>

typedef float v2f __attribute__((ext_vector_type(2)));
typedef float v8f __attribute__((ext_vector_type(8)));

#define DIM 1024
#define MROWS 64

// Each block: 256 threads = 8 waves (wave32).
// Wave w: mg = w&1 selects rows [mg*32, mg*32+32), ng = w>>1 selects a 32-col slice.
// Block covers all 64 rows x 128 columns of the output.
// Per K-step (K += 4): 4 x global_load_b64 (A0,A1,B0,B1 fragments) + 4 x v_wmma_f32_16x16x4_f32.
__global__ __launch_bounds__(256)
void sim_affine_relu_wmma(const float* __restrict__ claim, // [64][1024]
                          const float* __restrict__ emb,   // [N][1024]
                          const float* __restrict__ w,     // [N]
                          const float* __restrict__ bias,  // [N]
                          float* __restrict__ out,         // [64][N]
                          int N)
{
    const int lane = threadIdx.x & 31;
    const int wave = threadIdx.x >> 5;
    const int h    = lane >> 4;   // lane half: selects K pair {2h, 2h+1}
    const int r    = lane & 15;   // row-within-tile / col-within-tile
    const int mg   = wave & 1;    // 0..1 -> M rows mg*32 .. mg*32+31
    const int ng   = wave >> 1;   // 0..3 -> 32-column slice within block

    const long nbase = (long)blockIdx.x * 128 + (long)ng * 32;

    // A fragments: rows (mg*32 + r) and (mg*32 + 16 + r); per lane float2 at k + 2h
    const float* a0p = claim + ((long)(mg * 32 + r) * DIM) + 2 * h;
    const float* a1p = a0p + 16L * DIM;

    // B fragments: evidence rows nbase+r and nbase+16+r, clamped so every lane
    // issues a valid address (WMMA requires EXEC all-ones; OOB columns are
    // computed on clamped data and simply not stored).
    long n0 = nbase + r;        if (n0 > (long)N - 1) n0 = (long)N - 1;
    long n1 = nbase + 16 + r;   if (n1 > (long)N - 1) n1 = (long)N - 1;
    const float* b0p = emb + n0 * DIM + 2 * h;
    const float* b1p = emb + n1 * DIM + 2 * h;

    v8f acc00 = {}; v8f acc01 = {}; v8f acc10 = {}; v8f acc11 = {};

#pragma unroll 4
    for (int k = 0; k < DIM; k += 4) {
        v2f a0 = *(const v2f*)(a0p + k);
        v2f a1 = *(const v2f*)(a1p + k);
        v2f b0 = *(const v2f*)(b0p + k);
        v2f b1 = *(const v2f*)(b1p + k);
        acc00 = __builtin_amdgcn_wmma_f32_16x16x4_f32(false, a0, false, b0,
                                                      (short)0, acc00, false, false);
        acc01 = __builtin_amdgcn_wmma_f32_16x16x4_f32(false, a0, false, b1,
                                                      (short)0, acc01, false, false);
        acc10 = __builtin_amdgcn_wmma_f32_16x16x4_f32(false, a1, false, b0,
                                                      (short)0, acc10, false, false);
        acc11 = __builtin_amdgcn_wmma_f32_16x16x4_f32(false, a1, false, b1,
                                                      (short)0, acc11, false, false);
    }

    // Epilogue: relu(acc * w[n] + b[n]); predicated on true column index.
    const long nc0 = nbase + r;
    const long nc1 = nbase + 16 + r;
    const bool p0 = nc0 < (long)N;
    const bool p1 = nc1 < (long)N;
    const float w0  = p0 ? w[nc0]    : 0.0f;
    const float bb0 = p0 ? bias[nc0] : 0.0f;
    const float w1  = p1 ? w[nc1]    : 0.0f;
    const float bb1 = p1 ? bias[nc1] : 0.0f;

#pragma unroll
    for (int v = 0; v < 8; ++v) {
        // C/D layout: lane half h, VGPR v -> tile row (v + 8h)
        const int m0 = mg * 32 + v + 8 * h;       // M-tile 0 of this wave
        const int m1 = m0 + 16;                   // M-tile 1 of this wave
        if (p0) {
            out[(size_t)m0 * (size_t)N + (size_t)nc0] = fmaxf(acc00[v] * w0 + bb0, 0.0f);
            out[(size_t)m1 * (size_t)N + (size_t)nc0] = fmaxf(acc10[v] * w0 + bb0, 0.0f);
        }
        if (p1) {
            out[(size_t)m0 * (size_t)N + (size_t)nc1] = fmaxf(acc01[v] * w1 + bb1, 0.0f);
            out[(size_t)m1 * (size_t)N + (size_t)nc1] = fmaxf(acc11[v] * w1 + bb1, 0.0f);
        }
    }
}

extern "C" void kernel_launch(void* const* d_in, const int* in_sizes, int n_in,
                              void* d_out, int out_size, void* d_ws, size_t ws_size,
                              hipStream_t stream) {
    const float* claim     = (const float*)d_in[0];   // [64][1024]
    const float* text_emb  = (const float*)d_in[1];   // [N_text][1024]
    const float* image_emb = (const float*)d_in[2];   // [N_image][1024]
    const float* text_w    = (const float*)d_in[3];   // [N_text]
    const float* text_b    = (const float*)d_in[4];
    const float* image_w   = (const float*)d_in[5];   // [N_image]
    const float* image_b   = (const float*)d_in[6];

    const int n_text  = in_sizes[3];
    const int n_image = in_sizes[5];

    float* out_text  = (float*)d_out;
    float* out_image = out_text + (size_t)MROWS * (size_t)n_text;

    dim3 block(256);
    dim3 grid_text((n_text + 127) / 128);
    dim3 grid_image((n_image + 127) / 128);

    sim_affine_relu_wmma<<<grid_text, block, 0, stream>>>(
        claim, text_emb, text_w, text_b, out_text, n_text);
    sim_affine_relu_wmma<<<grid_image, block, 0, stream>>>(
        claim, image_emb, image_w, image_b, out_image, n_image);
}